// GCN_1layer_41807211659408
// MI455X (gfx1250) — compile-verified
//
#include <hip/hip_runtime.h>
#include <hip/hip_bf16.h>

#define F_IN   256
#define F_OUT  64

typedef float v2f __attribute__((ext_vector_type(2)));
typedef float v8f __attribute__((ext_vector_type(8)));

// ---------------------------------------------------------------------------
// Kernel 1: zero the aggregation buffer (d_out) — float4 stores.
// ---------------------------------------------------------------------------
__global__ void zero_f4_kernel(float4* __restrict__ p, int n4) {
    int i = blockIdx.x * blockDim.x + threadIdx.x;
    if (i < n4) p[i] = make_float4(0.f, 0.f, 0.f, 0.f);
}

// ---------------------------------------------------------------------------
// Kernel 2: support = x @ W via V_WMMA_F32_16X16X4_F32.
// Block = 256 threads = 8 waves; each wave owns one 16-row tile and all 64
// output columns (4 accumulators of 16 cols each). W is staged in LDS,
// transposed to [col][k] with an XOR swizzle (pk = k ^ ((col&15)<<2)) so the
// per-step ds_load_2addr_stride64_b64 hits 32 distinct banks.
// Epilogue: uniform in-range fast path (no per-lane exec juggling); guarded
// slow path only for a ragged final tile.
// ---------------------------------------------------------------------------
__global__ void __launch_bounds__(256)
gemm_wmma_f32_kernel(const float* __restrict__ x,
                     const float* __restrict__ Wg,
                     float* __restrict__ support,
                     int N) {
    __shared__ float wt[F_OUT * F_IN];   // 64 KB, swizzled transpose of W

    // Cooperative load + transpose + swizzle of W (one-time per block).
    for (int idx = threadIdx.x; idx < F_IN * F_OUT; idx += 256) {
        int k = idx >> 6;          // 0..255
        int c = idx & 63;          // 0..63
        wt[c * F_IN + (k ^ ((c & 15) << 2))] = Wg[idx];
    }
    __syncthreads();

    int waveId = threadIdx.x >> 5;
    int tile   = blockIdx.x * 8 + waveId;
    int numTiles = (N + 15) >> 4;
    if (tile >= numTiles) return;          // wave-uniform: EXEC stays all-1s

    int lane = threadIdx.x & 31;
    int half = lane >> 4;                  // 0: K={k0,k0+1}, 1: K={k0+2,k0+3}
    int m    = lane & 15;                  // row-in-tile (A) / col-in-ntile (B)

    int r = tile * 16 + m;
    const float* xrow = x + (size_t)((r < N) ? r : (N - 1)) * F_IN;

    v8f acc0 = {}, acc1 = {}, acc2 = {}, acc3 = {};

    #pragma unroll 4
    for (int k0 = 0; k0 < F_IN; k0 += 4) {
        int ka = k0 + (half << 1);
        v2f a = *(const v2f*)(xrow + ka);          // x[m][ka], x[m][ka+1]

        int pk = ka ^ (m << 2);                    // swizzled K index
        v2f b0 = *(const v2f*)&wt[(0 * 16 + m) * F_IN + pk];
        v2f b1 = *(const v2f*)&wt[(1 * 16 + m) * F_IN + pk];
        v2f b2 = *(const v2f*)&wt[(2 * 16 + m) * F_IN + pk];
        v2f b3 = *(const v2f*)&wt[(3 * 16 + m) * F_IN + pk];

        acc0 = __builtin_amdgcn_wmma_f32_16x16x4_f32(false, a, false, b0,
                                                     (short)0, acc0, false, false);
        acc1 = __builtin_amdgcn_wmma_f32_16x16x4_f32(false, a, false, b1,
                                                     (short)0, acc1, false, false);
        acc2 = __builtin_amdgcn_wmma_f32_16x16x4_f32(false, a, false, b2,
                                                     (short)0, acc2, false, false);
        acc3 = __builtin_amdgcn_wmma_f32_16x16x4_f32(false, a, false, b3,
                                                     (short)0, acc3, false, false);
    }

    // C/D layout: VGPR j -> M = j + 8*half, N = lane&15 (per 16-col tile).
    float* obase = support + (size_t)(tile * 16 + half * 8) * F_OUT + m;
    if (tile * 16 + 16 <= N) {
        // Uniform fast path: whole tile in range, no exec manipulation.
        #pragma unroll
        for (int j = 0; j < 8; j++) {
            float* o = obase + (size_t)j * F_OUT;
            o[ 0] = acc0[j];
            o[16] = acc1[j];
            o[32] = acc2[j];
            o[48] = acc3[j];
        }
    } else {
        #pragma unroll
        for (int j = 0; j < 8; j++) {
            if (tile * 16 + half * 8 + j < N) {
                float* o = obase + (size_t)j * F_OUT;
                o[ 0] = acc0[j];
                o[16] = acc1[j];
                o[32] = acc2[j];
                o[48] = acc3[j];
            }
        }
    }
}

// ---------------------------------------------------------------------------
// Kernel 3: edge scatter — msgs = edge_w * support[col]; agg[row] += msgs.
// One wave32 per edge, 2 features per lane. The edge index is wave-uniform;
// readfirstlane moves it to an SGPR so row/col/edge_w become scalar loads.
// Gather = two 128B coalesced reads; scatter = HW float atomics into the
// L2-resident agg (support+agg = 51MB inside the 192MB L2).
// ---------------------------------------------------------------------------
__global__ void __launch_bounds__(256)
edge_scatter_kernel(const int* __restrict__ row,
                    const int* __restrict__ col,
                    const float* __restrict__ ew,
                    const float* __restrict__ support,
                    float* __restrict__ agg,
                    int E) {
    int wid  = (blockIdx.x * blockDim.x + threadIdx.x) >> 5;  // edge (wave-uniform)
    int lane = threadIdx.x & 31;
    if (wid >= E) return;

#if defined(__HIP_DEVICE_COMPILE__)
    int e = __builtin_amdgcn_readfirstlane(wid);  // force SGPR -> scalar loads
#else
    int e = wid;
#endif
    int   c = col[e];
    int   r = row[e];
    float w = ew[e];

    const float* s = support + (size_t)c * F_OUT;
    float v0 = w * s[lane];
    float v1 = w * s[lane + 32];

    float* a = agg + (size_t)r * F_OUT;
#if defined(__HIP_DEVICE_COMPILE__)
    unsafeAtomicAdd(&a[lane],      v0);   // global_atomic_add_f32
    unsafeAtomicAdd(&a[lane + 32], v1);
#endif
}

// ---------------------------------------------------------------------------
// Kernel 4: h = relu(agg + b); out = log_softmax(h) — one wave32 per row,
// 2 features per lane, shuffle reductions. In-place on d_out.
// ---------------------------------------------------------------------------
__global__ void __launch_bounds__(256)
finalize_kernel(const float* __restrict__ bias,
                float* __restrict__ io,
                int N) {
    int gw   = (blockIdx.x * blockDim.x + threadIdx.x) >> 5;  // row
    int lane = threadIdx.x & 31;
    if (gw >= N) return;

    float* p = io + (size_t)gw * F_OUT;
    float h0 = p[lane]      + bias[lane];
    float h1 = p[lane + 32] + bias[lane + 32];
    h0 = fmaxf(h0, 0.f);
    h1 = fmaxf(h1, 0.f);

    float mx = fmaxf(h0, h1);
    #pragma unroll
    for (int off = 16; off > 0; off >>= 1)
        mx = fmaxf(mx, __shfl_xor(mx, off, 32));

    float s = expf(h0 - mx) + expf(h1 - mx);
    #pragma unroll
    for (int off = 16; off > 0; off >>= 1)
        s += __shfl_xor(s, off, 32);

    float ls = logf(s);
    p[lane]      = h0 - mx - ls;
    p[lane + 32] = h1 - mx - ls;
}

// ---------------------------------------------------------------------------
// Host-side launcher.
// Inputs (setup_inputs order): x[N,256] f32, row[E] i32, col[E] i32,
// edge_w[E] f32, W[256,64] f32, b[64] f32.  Output: [N,64] f32.
// Workspace: support[N,64] f32 (25.6 MB).
// ---------------------------------------------------------------------------
extern "C" void kernel_launch(void* const* d_in, const int* in_sizes, int n_in,
                              void* d_out, int out_size, void* d_ws, size_t ws_size,
                              hipStream_t stream) {
    const float* x    = (const float*)d_in[0];
    const int*   row  = (const int*)  d_in[1];
    const int*   col  = (const int*)  d_in[2];
    const float* ew   = (const float*)d_in[3];
    const float* W    = (const float*)d_in[4];
    const float* bias = (const float*)d_in[5];

    int N = in_sizes[0] / F_IN;
    int E = in_sizes[1];

    float* out     = (float*)d_out;   // doubles as agg accumulator
    float* support = (float*)d_ws;

    // 1) agg = 0
    int n4 = (N * F_OUT) / 4;
    zero_f4_kernel<<<(n4 + 255) / 256, 256, 0, stream>>>((float4*)out, n4);

    // 2) support = x @ W   (WMMA f32)
    int numTiles = (N + 15) >> 4;
    gemm_wmma_f32_kernel<<<(numTiles + 7) / 8, 256, 0, stream>>>(x, W, support, N);

    // 3) agg[row] += edge_w * support[col]  (one wave per edge)
    edge_scatter_kernel<<<(E + 7) / 8, 256, 0, stream>>>(row, col, ew, support, out, E);

    // 4) out = log_softmax(relu(agg + b))
    finalize_kernel<<<(N + 7) / 8, 256, 0, stream>>>(bias, out, N);
}